// Trilinear_85186381349185
// MI455X (gfx1250) — compile-verified
//
#include <hip/hip_runtime.h>

// CDNA5 / gfx1250 trilinear contraction:
//   out[b,a] = sum_k X[b,k] * W[a,k] + bias[a],  X[b,k]=x1[b,n]*x2[b,m]*x3[b,o]
// HBM-bound on streaming W (268 MB fp32, read once). fp32 WMMA 16x16x4 path,
// 32x16 C-tile per wave => 16 FLOP per W byte, matching the 23.3 TB/s roofline.
// v2: software-pipelined (double-buffered) W loads so 16 global_load_b64 stay
// in flight while 32 WMMAs execute, instead of load->wait0->wmma lockstep.

typedef float v2f __attribute__((ext_vector_type(2)));
typedef float v8f __attribute__((ext_vector_type(8)));

#define K_TOTAL 262144  // 64*64*64
#define OUT_A   256
#define BATCH   32

__device__ __forceinline__ void load_wgroup(const float* __restrict__ wrow,
                                            int nm, v2f (&buf)[16]) {
    const float* wg = wrow + (size_t)nm * 64;
#pragma unroll
    for (int j = 0; j < 16; ++j) buf[j] = *(const v2f*)(wg + j * 4);
}

__device__ __forceinline__ void load_p(const float* __restrict__ sx1,
                                       const float* __restrict__ sx2,
                                       int b0, int b1, int nm,
                                       float& p0, float& p1) {
    const int n = nm >> 6, m = nm & 63;
    p0 = sx1[b0 * 64 + n] * sx2[b0 * 64 + m];
    p1 = sx1[b1 * 64 + n] * sx2[b1 * 64 + m];
}

__device__ __forceinline__ void compute_group(float p0, float p1,
                                              const v2f (&wv)[16],
                                              const v2f (&x3a)[16],
                                              const v2f (&x3b)[16],
                                              v8f& c0, v8f& c1) {
#pragma unroll
    for (int j = 0; j < 16; ++j) {
        v2f a0; a0.x = p0 * x3a[j].x; a0.y = p0 * x3a[j].y;
        v2f a1; a1.x = p1 * x3b[j].x; a1.y = p1 * x3b[j].y;
        c0 = __builtin_amdgcn_wmma_f32_16x16x4_f32(
                 false, a0, false, wv[j], (short)0, c0, false, false);
        c1 = __builtin_amdgcn_wmma_f32_16x16x4_f32(
                 false, a1, false, wv[j], (short)0, c1, false, false);
    }
}

__global__ __launch_bounds__(256) void trilinear_wmma_stage1(
    const float* __restrict__ x1, const float* __restrict__ x2,
    const float* __restrict__ x3, const float* __restrict__ W,
    const float* __restrict__ bias, float* __restrict__ dst,
    int kPerChunk, int writeFinal)
{
    __shared__ float sx1[2048];   // 32 x 64
    __shared__ float sx2[2048];
    __shared__ float sx3[2048];
    __shared__ float sred[4096];  // 8 waves x 512 C elements

    const int tid = threadIdx.x;
    for (int i = tid; i < 2048; i += 256) {
        sx1[i] = x1[i]; sx2[i] = x2[i]; sx3[i] = x3[i];
    }
    __syncthreads();

    const int lane  = tid & 31;
    const int wave  = tid >> 5;
    const int aBase = blockIdx.x * 16;
    const int chunk = blockIdx.y;
    const int kw     = kPerChunk >> 3;               // K per wave (multiple of 64)
    const int kStart = chunk * kPerChunk + wave * kw;
    const int nGroups = kw >> 6;                     // 64-k groups; always even

    const int bl  = lane & 15;           // M index within tile / N index for B
    const int klo = (lane >> 4) << 1;    // K sub-offset per ISA A/B fp32 layout: 0 or 2
    const int b0 = bl, b1 = bl + 16;     // two batch rows per lane (two M-tiles)

    // Hoist per-lane x3 pairs: k-pattern within a 64-group is invariant across groups.
    v2f x3a[16], x3b[16];
#pragma unroll
    for (int j = 0; j < 16; ++j) {
        const int o = j * 4 + klo;
        x3a[j] = *(const v2f*)&sx3[b0 * 64 + o];
        x3b[j] = *(const v2f*)&sx3[b1 * 64 + o];
    }

    // B operand: lane holds column a = aBase + (lane&15), K offset klo. Row stride 1 MB.
    const float* wrow = W + (size_t)(aBase + bl) * K_TOTAL + klo;

    v8f c0 = {}; // b = 0..15  x a-tile
    v8f c1 = {}; // b = 16..31 x a-tile

    const int gBase = kStart >> 6;
    v2f wA[16], wB[16];
    float p0A, p1A, p0B, p1B;

    // Prologue: prefetch group 0.
    load_wgroup(wrow, gBase, wA);
    load_p(sx1, sx2, b0, b1, gBase, p0A, p1A);

    for (int g = 0; g < nGroups; g += 2) {
        // Prefetch group g+1 while computing group g.
        const int nm1 = (g + 1 < nGroups) ? (gBase + g + 1) : gBase;  // clamp (safety)
        load_wgroup(wrow, nm1, wB);
        load_p(sx1, sx2, b0, b1, nm1, p0B, p1B);
        compute_group(p0A, p1A, wA, x3a, x3b, c0, c1);

        // Prefetch group g+2 while computing group g+1.
        const int nm2 = (g + 2 < nGroups) ? (gBase + g + 2) : gBase;  // clamp at tail
        load_wgroup(wrow, nm2, wA);
        load_p(sx1, sx2, b0, b1, nm2, p0A, p1A);
        compute_group(p0B, p1B, wB, x3a, x3b, c0, c1);
    }

    // Deterministic cross-wave reduction of the 8 partial 32x16 tiles via LDS.
    __syncthreads();
#pragma unroll
    for (int r = 0; r < 8; ++r) {
        sred[wave * 512 +       r * 32 + lane] = c0[r];
        sred[wave * 512 + 256 + r * 32 + lane] = c1[r];
    }
    __syncthreads();

    for (int t = tid; t < 512; t += 256) {
        float acc = 0.f;
#pragma unroll
        for (int w = 0; w < 8; ++w) acc += sred[w * 512 + t];
        // Decode C layout: VGPR r, lane ln -> M = r + (ln>=16 ? 8 : 0), N = ln&15.
        const int tile = t >> 8;
        const int rr   = (t >> 5) & 7;
        const int ln   = t & 31;
        const int b = tile * 16 + rr + ((ln >= 16) ? 8 : 0);
        const int a = aBase + (ln & 15);
        const int idx = b * OUT_A + a;
        if (writeFinal) dst[idx] = acc + bias[a];
        else            dst[(size_t)chunk * (BATCH * OUT_A) + idx] = acc;
    }
}

__global__ __launch_bounds__(256) void trilinear_stage2(
    const float* __restrict__ partial, const float* __restrict__ bias,
    float* __restrict__ out, int chunks)
{
    const int idx = blockIdx.x * 256 + threadIdx.x;  // 8192 = 32*256 outputs
    float acc = bias[idx & (OUT_A - 1)];
    for (int c = 0; c < chunks; ++c)
        acc += partial[(size_t)c * (BATCH * OUT_A) + idx];
    out[idx] = acc;
}

extern "C" void kernel_launch(void* const* d_in, const int* in_sizes, int n_in,
                              void* d_out, int out_size, void* d_ws, size_t ws_size,
                              hipStream_t stream) {
    const float* x1   = (const float*)d_in[0];
    const float* x2   = (const float*)d_in[1];
    const float* x3   = (const float*)d_in[2];
    const float* W    = (const float*)d_in[3];
    const float* bias = (const float*)d_in[4];
    float* out = (float*)d_out;

    // Pick largest power-of-two K-chunk count (<=64) whose partials fit in ws.
    const size_t partialBytes = (size_t)BATCH * OUT_A * sizeof(float); // 32 KB
    int chunks = 64;
    while (chunks > 1 && (size_t)chunks * partialBytes > ws_size) chunks >>= 1;

    const int kPerChunk = K_TOTAL / chunks;
    const int writeFinal = (chunks == 1) ? 1 : 0;
    float* dst = writeFinal ? out : (float*)d_ws;

    dim3 grid(OUT_A / 16, chunks);
    trilinear_wmma_stage1<<<grid, 256, 0, stream>>>(
        x1, x2, x3, W, bias, dst, kPerChunk, writeFinal);

    if (!writeFinal) {
        trilinear_stage2<<<(BATCH * OUT_A) / 256, 256, 0, stream>>>(
            (const float*)d_ws, bias, out, chunks);
    }
}